// GraphAutoencoderTra_43997644980277
// MI455X (gfx1250) — compile-verified
//
#include <hip/hip_runtime.h>

// ---------------- problem constants (match reference) ----------------
#define NN   100000    // nodes
#define CC   512       // structural communities
#define FF   64        // functional communities
#define HH   128       // hidden dim
#define EE   1600000   // adjacency edges
#define ESS  1000000   // scored edges
#define NP   112       // 100 output cols padded to 7*16

typedef __attribute__((ext_vector_type(2))) float v2f;
typedef __attribute__((ext_vector_type(8))) float v8f;

// D = A(16x4) * B(4x16) + C, f32 WMMA (CDNA5 native f32 matrix op)
__device__ __forceinline__ v8f wmma4(v2f a, v2f b, v8f c) {
  return __builtin_amdgcn_wmma_f32_16x16x4_f32(false, a, false, b, (short)0, c,
                                               false, false);
}

// 8-byte-aligned paired load (callers guarantee alignment)
__device__ __forceinline__ v2f ld2(const float* p) {
  return *(const v2f*)p;
}

// ---------------- utility kernels ----------------
__global__ void k_zero(float* p, long long n) {
  long long i = (long long)blockIdx.x * blockDim.x + threadIdx.x;
  long long s = (long long)gridDim.x * blockDim.x;
  for (; i < n; i += s) p[i] = 0.f;
}

// column sums of SA [NN, CC]; 256 threads, each owns cols t and t+256
__global__ void k_colsum(const float* __restrict__ SA, float* colsum) {
  int t = threadIdx.x;
  int rows_per = (NN + gridDim.x - 1) / gridDim.x;
  int n0 = blockIdx.x * rows_per;
  int n1 = n0 + rows_per; if (n1 > NN) n1 = NN;
  float a0 = 0.f, a1 = 0.f;
  const float* row = SA + (long long)n0 * CC;
  for (int n = n0; n < n1; ++n, row += CC) {
    a0 += row[t];
    a1 += row[t + 256];
  }
  atomicAdd(&colsum[t], a0);
  atomicAdd(&colsum[t + 256], a1);
}

__global__ void k_inv(const float* colsum, float* inv) {
  int c = blockIdx.x * blockDim.x + threadIdx.x;
  if (c < CC) {
    float cs = colsum[c] - 1.f;
    if (cs < 0.f) cs = 0.f;
    inv[c] = 1.f / (cs + 1.f);
  }
}

// zero-padded copies: lin_W [384,100] -> [384,112], lin_b [100] -> [112]
__global__ void k_padW(const float* __restrict__ W, const float* __restrict__ b,
                       float* __restrict__ Wp, float* __restrict__ bp) {
  int idx = blockIdx.x * blockDim.x + threadIdx.x;
  if (idx < 384 * NP) {
    int r = idx / NP, c = idx - r * NP;
    Wp[idx] = (c < 100) ? W[r * 100 + c] : 0.f;
  }
  if (idx < NP) bp[idx] = (idx < 100) ? b[idx] : 0.f;
}

// raw_feat[n, 0:128] = concat of 4 embedding gathers (each 32 wide)
__global__ void k_rawfeat(const int* __restrict__ lane, const int* __restrict__ type,
                          const int* __restrict__ len, const int* __restrict__ node,
                          const float* __restrict__ laneW, const float* __restrict__ typeW,
                          const float* __restrict__ lenW, const float* __restrict__ nodeW,
                          float* __restrict__ rf) {
  long long idx = (long long)blockIdx.x * blockDim.x + threadIdx.x;
  if (idx >= (long long)NN * HH) return;
  int n = (int)(idx >> 7);
  int h = (int)(idx & 127);
  float v;
  if (h < 32)      v = laneW[lane[n] * 32 + h];
  else if (h < 64) v = typeW[type[n] * 32 + h - 32];
  else if (h < 96) v = lenW[len[n] * 32 + h - 64];
  else             v = nodeW[node[n] * 32 + h - 96];
  rf[idx] = v;
}

// struct_inter[src[e], :] += SA[dst[e], :]  (unit-valued sparse A times SA)
__global__ void k_scatter(const int* __restrict__ adj, const float* __restrict__ SA,
                          float* __restrict__ SI) {
  long long idx = (long long)blockIdx.x * blockDim.x + threadIdx.x;
  if (idx >= (long long)EE * 128) return;
  int e = (int)(idx >> 7);
  int i = (int)(idx & 127);       // 4 columns per thread
  int src = adj[e];
  int dst = adj[EE + e];
  const float4 v = *(const float4*)(SA + (long long)dst * CC + i * 4);
  float* o = SI + (long long)src * CC + i * 4;
  atomicAdd(o + 0, v.x); atomicAdd(o + 1, v.y);
  atomicAdd(o + 2, v.z); atomicAdd(o + 3, v.w);
}

// ---------------- WMMA GEMM kernels ----------------
// C[m, n..] += sum_k A[k*lda + m] * B[k*ldb + n]  (A accessed transposed)
// NT output column tiles per wave (A-fragment reuse).
// grid = (Ntiles/NT, Mtiles, Ksplit); block = 32 (one wave)
template <int NT>
__global__ void k_gemm_at(const float* __restrict__ A, int lda,
                          const float* __restrict__ B, int ldb,
                          float* __restrict__ Cm, int ldc,
                          int kchunk, int Ktot) {
  int lane = threadIdx.x & 31;
  int lm = lane & 15;
  int half = lane >> 4;
  int n0 = blockIdx.x * (16 * NT);
  int m0 = blockIdx.y * 16;
  int kb = blockIdx.z * kchunk;
  int ke = kb + kchunk; if (ke > Ktot) ke = Ktot;
  v8f acc[NT];
#pragma unroll
  for (int t = 0; t < NT; ++t) acc[t] = (v8f){0.f,0.f,0.f,0.f,0.f,0.f,0.f,0.f};

  const float* pa = A + (long long)(kb + 2 * half) * lda + m0 + lm;
  const float* pb = B + (long long)(kb + 2 * half) * ldb + n0 + lm;
  const long long stepa = 4LL * lda;
  const long long stepb = 4LL * ldb;
  for (int k = kb; k < ke; k += 4) {
    __builtin_prefetch(pa + 4 * stepa, 0, 1);   // 16 K ahead on strided A walk
    __builtin_prefetch(pb + 4 * stepb, 0, 1);
    v2f a; a.x = pa[0]; a.y = pa[lda];
#pragma unroll
    for (int t = 0; t < NT; ++t) {
      v2f b; b.x = pb[t * 16]; b.y = pb[ldb + t * 16];
      acc[t] = wmma4(a, b, acc[t]);
    }
    pa += stepa; pb += stepb;
  }
#pragma unroll
  for (int t = 0; t < NT; ++t) {
#pragma unroll
    for (int v = 0; v < 8; ++v) {
      int m = m0 + v + 8 * half;
      atomicAdd(&Cm[(long long)m * ldc + n0 + t * 16 + lm], acc[t][v]);
    }
  }
}

// C[m,n] (+)= sum_k A[m*lda + k] * B[k*ldb + n]; all dims multiples of 16,
// so no guards -> no exec-mask branches in the inner loop.
// grid = (Ntiles, Mtiles); block = 32
__global__ void k_gemm_a(const float* __restrict__ A, int lda,
                         const float* __restrict__ B, int ldb,
                         float* __restrict__ Cm, int ldc,
                         int Ktot, int accumulate) {
  int lane = threadIdx.x & 31;
  int lm = lane & 15;
  int half = lane >> 4;
  int n0 = blockIdx.x * 16;
  int m0 = blockIdx.y * 16;
  v8f acc = {0.f, 0.f, 0.f, 0.f, 0.f, 0.f, 0.f, 0.f};
  const float* pa = A + (long long)(m0 + lm) * lda + 2 * half;   // 8B aligned
  const float* pb = B + (long long)(2 * half) * ldb + n0 + lm;
  const long long stepb = 4LL * ldb;
  for (int k = 0; k < Ktot; k += 4) {
    v2f a = ld2(pa);
    v2f b;
    b.x = pb[0];
    b.y = pb[ldb];
    acc = wmma4(a, b, acc);
    pa += 4; pb += stepb;
  }
#pragma unroll
  for (int v = 0; v < 8; ++v) {
    int m = m0 + v + 8 * half;
    if (accumulate) atomicAdd(&Cm[(long long)m * ldc + n0 + lm], acc[v]);
    else            Cm[(long long)m * ldc + n0 + lm] = acc[v];
  }
}

// ---------------- small elementwise kernels ----------------
__global__ void k_scale_rows(float* M, const float* inv, int cols, int total) {
  int idx = blockIdx.x * blockDim.x + threadIdx.x;
  if (idx < total) M[idx] *= inv[idx / cols];
}

__global__ void k_scale_adj(float* adj, const float* inv) {
  int idx = blockIdx.x * blockDim.x + threadIdx.x;
  if (idx < CC * CC) {
    int r = idx >> 9, c = idx & 511;
    adj[idx] *= inv[r] * inv[c];
  }
}

// softmax over rows (axis 0) of fa [CC, FF], one block per column f;
// also writes fa_s[c,f] = softmax * inv[c]
__global__ void k_softmax_scale(float* __restrict__ fa, float* __restrict__ fa_s,
                                const float* __restrict__ inv) {
  __shared__ float red[256];
  int f = blockIdx.x;
  int t = threadIdx.x;
  float v0 = fa[t * FF + f];
  float v1 = fa[(t + 256) * FF + f];
  red[t] = fmaxf(v0, v1);
  __syncthreads();
  for (int s = 128; s > 0; s >>= 1) {
    if (t < s) red[t] = fmaxf(red[t], red[t + s]);
    __syncthreads();
  }
  float mx = red[0];
  __syncthreads();
  float e0 = __expf(v0 - mx), e1 = __expf(v1 - mx);
  red[t] = e0 + e1;
  __syncthreads();
  for (int s = 128; s > 0; s >>= 1) {
    if (t < s) red[t] += red[t + s];
    __syncthreads();
  }
  float isum = 1.f / red[0];
  float r0 = e0 * isum, r1 = e1 * isum;
  fa[t * FF + f] = r0;
  fa[(t + 256) * FF + f] = r1;
  fa_s[t * FF + f] = r0 * inv[t];
  fa_s[(t + 256) * FF + f] = r1 * inv[t + 256];
}

// ---------------- fused final linear: Xp = rf@W0p + SA@Gp + bp ----------------
// 8 waves/block, one 16-row tile per wave, 7 column tiles (padded, no guards)
__global__ void k_X(const float* __restrict__ rf, const float* __restrict__ SA,
                    const float* __restrict__ W0p, const float* __restrict__ Gp,
                    const float* __restrict__ bp, float* __restrict__ Xp) {
  int lane = threadIdx.x & 31;
  int wv = threadIdx.x >> 5;
  int mtile = blockIdx.x * 8 + wv;
  if (mtile >= NN / 16) return;   // wave-uniform exit
  int lm = lane & 15, half = lane >> 4;
  int m0 = mtile * 16;
  v8f acc[7];
#pragma unroll
  for (int nt = 0; nt < 7; ++nt)
    acc[nt] = (v8f){0.f, 0.f, 0.f, 0.f, 0.f, 0.f, 0.f, 0.f};
  // phase 1: raw_feat @ W0p (K = 128)
  {
    const float* pa = rf + (long long)(m0 + lm) * HH + 2 * half;   // 8B aligned
    const float* pb = W0p + (2 * half) * NP + lm;
    for (int k = 0; k < HH; k += 4) {
      v2f a = ld2(pa);
#pragma unroll
      for (int nt = 0; nt < 7; ++nt) {
        v2f b;
        b.x = pb[nt * 16];
        b.y = pb[NP + nt * 16];
        acc[nt] = wmma4(a, b, acc[nt]);
      }
      pa += 4; pb += 4 * NP;
    }
  }
  // phase 2: SA @ Gp (K = 512)
  {
    const float* pa = SA + (long long)(m0 + lm) * CC + 2 * half;   // 8B aligned
    const float* pb = Gp + (2 * half) * NP + lm;
    for (int k = 0; k < CC; k += 4) {
      v2f a = ld2(pa);
#pragma unroll
      for (int nt = 0; nt < 7; ++nt) {
        v2f b;
        b.x = pb[nt * 16];
        b.y = pb[NP + nt * 16];
        acc[nt] = wmma4(a, b, acc[nt]);
      }
      pa += 4; pb += 4 * NP;
    }
  }
  // store with bias (full padded width, no guards; pad cols are don't-care)
#pragma unroll
  for (int nt = 0; nt < 7; ++nt) {
    int n = nt * 16 + lm;
    float bb = bp[n];
#pragma unroll
    for (int v = 0; v < 8; ++v) {
      int m = m0 + v + 8 * half;
      Xp[(long long)m * NP + n] = acc[nt][v] + bb;
    }
  }
}

// pred[e] = dot(Xp[s0[e]], Xp[s1[e]]) over first 100 of 112 cols; wave/edge
__global__ void k_edge(const int* __restrict__ se, const float* __restrict__ Xp,
                       float* __restrict__ out) {
  int w = (int)(((long long)blockIdx.x * blockDim.x + threadIdx.x) >> 5);
  int lane = threadIdx.x & 31;
  if (w >= ESS) return;
  const float* x0 = Xp + (long long)se[w] * NP;
  const float* x1 = Xp + (long long)se[ESS + w] * NP;
  float p = 0.f;
  for (int c = lane; c < 100; c += 32) p += x0[c] * x1[c];
  for (int off = 16; off > 0; off >>= 1) p += __shfl_xor(p, off, 32);
  if (lane == 0) out[w] = p;
}

// ---------------- workspace layout (floats) ----------------
// zero-initialized region first (atomicAdd targets), then plain buffers
#define OFF_COLSUM  0LL
#define OFF_INV     512LL
#define OFF_SE      1024LL                        // 512*128
#define OFF_ADJ     (OFF_SE + 512LL * 128)        // 512*512
#define OFF_FEMB    (OFF_ADJ + 512LL * 512)       // 64*128
#define OFF_G       (OFF_FEMB + 64LL * 128)       // 512*112 (padded)
#define OFF_SI      (OFF_G + 512LL * NP)          // NN*512
#define ZERO_COUNT  (OFF_SI + (long long)NN * CC)
#define OFF_RF      ZERO_COUNT                    // NN*128
#define OFF_HID     (OFF_RF + (long long)NN * HH) // 512*64
#define OFF_FA      (OFF_HID + 512LL * 64)        // 512*64
#define OFF_FAS     (OFF_FA + 512LL * 64)         // 512*64
#define OFF_M2      (OFF_FAS + 512LL * 64)        // 512*128
#define OFF_WP      (OFF_M2 + 512LL * 128)        // 384*112 (padded lin_W)
#define OFF_BP      (OFF_WP + 384LL * NP)         // 112 (padded lin_b)
#define OFF_X       (OFF_BP + NP)                 // NN*112 (padded X)
// total ~75.8M floats (~304 MB) required in d_ws

extern "C" void kernel_launch(void* const* d_in, const int* in_sizes, int n_in,
                              void* d_out, int out_size, void* d_ws, size_t ws_size,
                              hipStream_t stream) {
  (void)in_sizes; (void)n_in; (void)out_size; (void)ws_size;
  const int*   lane  = (const int*)d_in[0];
  const int*   type  = (const int*)d_in[1];
  const int*   len   = (const int*)d_in[2];
  const int*   node  = (const int*)d_in[3];
  const int*   adjI  = (const int*)d_in[4];
  const float* SA    = (const float*)d_in[6];
  const int*   se    = (const int*)d_in[7];
  const float* laneW = (const float*)d_in[8];
  const float* typeW = (const float*)d_in[9];
  const float* lenW  = (const float*)d_in[10];
  const float* nodeW = (const float*)d_in[11];
  const float* gcnW  = (const float*)d_in[12];
  const float* linW  = (const float*)d_in[13];
  const float* linB  = (const float*)d_in[14];
  float* out = (float*)d_out;

  float* ws     = (float*)d_ws;
  float* colsum = ws + OFF_COLSUM;
  float* inv    = ws + OFF_INV;
  float* SE     = ws + OFF_SE;    // struct_emb (scaled in place)
  float* adj    = ws + OFF_ADJ;   // struct_adj (scaled in place)
  float* femb   = ws + OFF_FEMB;  // fnc_emb
  float* Gp     = ws + OFF_G;     // padded G [512,112]
  float* SI     = ws + OFF_SI;    // struct_inter
  float* rf     = ws + OFF_RF;    // raw_feat
  float* hid    = ws + OFF_HID;   // struct_emb @ gcn_W
  float* fa     = ws + OFF_FA;    // fnc_assign
  float* fas    = ws + OFF_FAS;   // inv-scaled fnc_assign
  float* M2     = ws + OFF_M2;    // fa_s @ fnc_emb
  float* Wp     = ws + OFF_WP;    // padded lin_W [384,112]
  float* bp     = ws + OFF_BP;    // padded lin_b [112]
  float* Xp     = ws + OFF_X;     // padded X [NN,112]

  const float* W0p = Wp;                // [128,112]
  const float* W1p = Wp + 128 * NP;     // [128,112]
  const float* W2p = Wp + 256 * NP;     // [128,112]

  // 1) zero atomic-accumulation buffers + struct_inter
  k_zero<<<4096, 256, 0, stream>>>(ws, ZERO_COUNT);
  // 2) column sums -> inv scale; pad lin_W/lin_b
  k_colsum<<<512, 256, 0, stream>>>(SA, colsum);
  k_inv<<<2, 256, 0, stream>>>(colsum, inv);
  k_padW<<<(384 * NP + 255) / 256, 256, 0, stream>>>(linW, linB, Wp, bp);
  // 3) raw_feat gather
  k_rawfeat<<<(int)(((long long)NN * HH + 255) / 256), 256, 0, stream>>>(
      lane, type, len, node, laneW, typeW, lenW, nodeW, rf);
  // 4) struct_inter = A @ SA (edge scatter-add)
  k_scatter<<<(int)(((long long)EE * 128 + 255) / 256), 256, 0, stream>>>(adjI, SA, SI);
  // 5) SE0 = SA^T @ raw_feat  (M=512, N=128, K=100000, split-K 50, NT=4)
  k_gemm_at<4><<<dim3(2, 32, 50), 32, 0, stream>>>(SA, CC, rf, HH, SE, HH, 2000, NN);
  // 6) SE *= inv[row]   (struct_emb)
  k_scale_rows<<<(512 * 128 + 255) / 256, 256, 0, stream>>>(SE, inv, HH, 512 * 128);
  // 7) adj0 = SA^T @ SI  (M=512, N=512, K=100000, split-K 50, NT=4)
  k_gemm_at<4><<<dim3(8, 32, 50), 32, 0, stream>>>(SA, CC, SI, CC, adj, CC, 2000, NN);
  // 8) adj *= inv[r]*inv[c]  (struct_adj)
  k_scale_adj<<<(512 * 512 + 255) / 256, 256, 0, stream>>>(adj, inv);
  // 9) hid = SE @ gcn_W   (512x128x64)
  k_gemm_a<<<dim3(4, 32), 32, 0, stream>>>(SE, HH, gcnW, FF, hid, FF, HH, 0);
  // 10) fa = adj @ hid    (512x512x64)
  k_gemm_a<<<dim3(4, 32), 32, 0, stream>>>(adj, CC, hid, FF, fa, FF, CC, 0);
  // 11) softmax over axis 0 + fa_s = fa * inv[row]
  k_softmax_scale<<<FF, 256, 0, stream>>>(fa, fas, inv);
  // 12) fnc_emb = fa^T @ SE   (64x512x128, transposed-A, NT=4)
  k_gemm_at<4><<<dim3(2, 4, 1), 32, 0, stream>>>(fa, FF, SE, HH, femb, HH, CC, CC);
  // 13) M2 = fa_s @ fnc_emb   (512x64x128)
  k_gemm_a<<<dim3(8, 32), 32, 0, stream>>>(fas, FF, femb, HH, M2, HH, FF, 0);
  // 14) Gp = SE @ W1p + M2 @ W2p   (512x128x112, accumulate into zeroed Gp)
  k_gemm_a<<<dim3(7, 32), 32, 0, stream>>>(SE, HH, W1p, NP, Gp, NP, HH, 1);
  k_gemm_a<<<dim3(7, 32), 32, 0, stream>>>(M2, HH, W2p, NP, Gp, NP, HH, 1);
  // 15) Xp = rf @ W0p + SA @ Gp + bp
  k_X<<<(NN / 16 + 7) / 8, 256, 0, stream>>>(rf, SA, W0p, Gp, bp, Xp);
  // 16) pred_edge[e] = dot(Xp[s0], Xp[s1]) over first 100 cols
  k_edge<<<(int)(((long long)ESS * 32 + 255) / 256), 256, 0, stream>>>(se, Xp, out);
}